// DeltaLag_52725018525727
// MI455X (gfx1250) — compile-verified
//
#include <hip/hip_runtime.h>
#include <math.h>

// ---------------------------------------------------------------------------
// DeltaLag pipeline for MI455X (gfx1250, wave32, WMMA).
// fp32 end-to-end using V_WMMA_F32_16X16X4_F32 (fp32 A/B, fp32 acc).
// ---------------------------------------------------------------------------

typedef float v2f __attribute__((ext_vector_type(2)));
typedef float v8f __attribute__((ext_vector_type(8)));

#define N_STOCKS 2048
#define T_STEPS  64
#define F_DIM    16
#define H_DIM    64
#define L_LAGS   16
#define TOPK_K   10
#define G3       192   // 3*H

__device__ __forceinline__ v8f wmma4(v2f a, v2f b, v8f c) {
  // D = A(16x4,f32) * B(4x16,f32) + C(16x16,f32)
  return __builtin_amdgcn_wmma_f32_16x16x4_f32(
      /*neg_a=*/false, a, /*neg_b=*/false, b,
      /*c_mod=*/(short)0, c, /*reuse_a=*/false, /*reuse_b=*/false);
}

__device__ __forceinline__ float sigmoidf_(float v) {
  return 1.0f / (1.0f + __expf(-v));
}

// ---------------------------------------------------------------------------
// Kernel 1: fused GRU. One block = 16 samples, 4 waves; wave w owns hidden
// columns [16w,16w+16). Per step: 12 WMMA (x-gates, K=16) + 48 WMMA
// (h-gates, K=64). h lives in LDS; weight B-fragments hoisted to registers.
// Writes h for the last L=16 steps to hkeys (N, L, H).
// ---------------------------------------------------------------------------
__global__ __launch_bounds__(128) void DeltaLag_gru(
    const float* __restrict__ x,     // (N,T,F)
    const float* __restrict__ W_ih,  // (3H,F)
    const float* __restrict__ W_hh,  // (3H,H)
    const float* __restrict__ b_ih,  // (3H)
    const float* __restrict__ b_hh,  // (3H)
    float* __restrict__ hkeys)       // (N,L,H)
{
  __shared__ float hbuf[16 * H_DIM];

  const int tid  = threadIdx.x;
  const int wave = tid >> 5;
  const int lane = tid & 31;
  const int m    = lane & 15;   // A row / B col within 16x16 tile
  const int khi  = lane >> 4;   // k sub-offset selector (0/1)
  const int n0   = blockIdx.x * 16;

  for (int i = tid; i < 16 * H_DIM; i += 128) hbuf[i] = 0.0f;

  const int jr = 16 * wave + m;        // gate column (r)
  const int jz = 16 * (wave + 4) + m;  // gate column (z)
  const int jn = 16 * (wave + 8) + m;  // gate column (n)

  // t-invariant weight B-fragments -> registers
  v2f wr[4], wz[4], wn[4];
#pragma unroll
  for (int s = 0; s < 4; ++s) {
    int kk = 4 * s + 2 * khi;
    wr[s].x = W_ih[jr * F_DIM + kk]; wr[s].y = W_ih[jr * F_DIM + kk + 1];
    wz[s].x = W_ih[jz * F_DIM + kk]; wz[s].y = W_ih[jz * F_DIM + kk + 1];
    wn[s].x = W_ih[jn * F_DIM + kk]; wn[s].y = W_ih[jn * F_DIM + kk + 1];
  }
  v2f ur[16], uz[16], un[16];
#pragma unroll
  for (int s = 0; s < 16; ++s) {
    int kk = 4 * s + 2 * khi;
    ur[s].x = W_hh[jr * H_DIM + kk]; ur[s].y = W_hh[jr * H_DIM + kk + 1];
    uz[s].x = W_hh[jz * H_DIM + kk]; uz[s].y = W_hh[jz * H_DIM + kk + 1];
    un[s].x = W_hh[jn * H_DIM + kk]; un[s].y = W_hh[jn * H_DIM + kk + 1];
  }
  const float bxr = b_ih[jr], bxz = b_ih[jz], bxn = b_ih[jn];
  const float bhr = b_hh[jr], bhz = b_hh[jz], bhn = b_hh[jn];
  const int hc = 16 * wave + m;  // h column covered by this lane's C tiles

  const float* xrow = x + (size_t)(n0 + m) * (T_STEPS * F_DIM);

  __syncthreads();

#pragma unroll 1
  for (int t = 0; t < T_STEPS; ++t) {
    v8f aXr = {}, aXz = {}, aXn = {}, aHr = {}, aHz = {}, aHn = {};
    // x-gates: K = F = 16 -> 4 slices
#pragma unroll
    for (int s = 0; s < 4; ++s) {
      int kk = 4 * s + 2 * khi;
      v2f a; a.x = xrow[t * F_DIM + kk]; a.y = xrow[t * F_DIM + kk + 1];
      aXr = wmma4(a, wr[s], aXr);
      aXz = wmma4(a, wz[s], aXz);
      aXn = wmma4(a, wn[s], aXn);
    }
    // h-gates: K = H = 64 -> 16 slices, A from LDS
#pragma unroll
    for (int s = 0; s < 16; ++s) {
      int kk = 4 * s + 2 * khi;
      v2f a; a.x = hbuf[m * H_DIM + kk]; a.y = hbuf[m * H_DIM + kk + 1];
      aHr = wmma4(a, ur[s], aHr);
      aHz = wmma4(a, uz[s], aHz);
      aHn = wmma4(a, un[s], aHn);
    }
    __syncthreads();  // everyone is done reading hbuf for this step
#pragma unroll
    for (int r = 0; r < 8; ++r) {
      int row = r + 8 * khi;
      float hold = hbuf[row * H_DIM + hc];
      float rg = sigmoidf_(aXr[r] + bxr + aHr[r] + bhr);
      float zg = sigmoidf_(aXz[r] + bxz + aHz[r] + bhz);
      float ng = tanhf(aXn[r] + bxn + rg * (aHn[r] + bhn));
      float hnew = (1.0f - zg) * ng + zg * hold;
      hbuf[row * H_DIM + hc] = hnew;
      if (t >= T_STEPS - L_LAGS) {
        hkeys[(size_t)(n0 + row) * (L_LAGS * H_DIM) +
              (t - (T_STEPS - L_LAGS)) * H_DIM + hc] = hnew;
      }
    }
    __syncthreads();  // hbuf fully updated before next step's reads
  }
}

// ---------------------------------------------------------------------------
// Kernel 2: out(Mx64) = A(Mx64) @ W(64x64)^T. One wave per 16 rows;
// 4 column tiles x 16 k-slices = 64 WMMAs. Used for both keys and queries
// (queries use strideA=1024 to pick h[:,T-1] out of hkeys).
// ---------------------------------------------------------------------------
__global__ __launch_bounds__(32) void DeltaLag_proj(
    const float* __restrict__ A, int strideA,
    const float* __restrict__ W, float* __restrict__ out)
{
  const int lane = threadIdx.x;
  const int m = lane & 15, khi = lane >> 4;
  const int row0 = blockIdx.x * 16;

  v2f wf0[16], wf1[16], wf2[16], wf3[16];
#pragma unroll
  for (int s = 0; s < 16; ++s) {
    int kk = 4 * s + 2 * khi;
    wf0[s].x = W[(m)*H_DIM + kk];        wf0[s].y = W[(m)*H_DIM + kk + 1];
    wf1[s].x = W[(16 + m)*H_DIM + kk];   wf1[s].y = W[(16 + m)*H_DIM + kk + 1];
    wf2[s].x = W[(32 + m)*H_DIM + kk];   wf2[s].y = W[(32 + m)*H_DIM + kk + 1];
    wf3[s].x = W[(48 + m)*H_DIM + kk];   wf3[s].y = W[(48 + m)*H_DIM + kk + 1];
  }
  const float* arow = A + (size_t)(row0 + m) * strideA;
  v8f acc0 = {}, acc1 = {}, acc2 = {}, acc3 = {};
#pragma unroll
  for (int s = 0; s < 16; ++s) {
    int kk = 4 * s + 2 * khi;
    v2f a; a.x = arow[kk]; a.y = arow[kk + 1];
    acc0 = wmma4(a, wf0[s], acc0);
    acc1 = wmma4(a, wf1[s], acc1);
    acc2 = wmma4(a, wf2[s], acc2);
    acc3 = wmma4(a, wf3[s], acc3);
  }
#pragma unroll
  for (int r = 0; r < 8; ++r) {
    size_t row = (size_t)(row0 + r + 8 * khi);
    out[row * H_DIM + m]      = acc0[r];
    out[row * H_DIM + 16 + m] = acc1[r];
    out[row * H_DIM + 32 + m] = acc2[r];
    out[row * H_DIM + 48 + m] = acc3[r];
  }
}

// ---------------------------------------------------------------------------
// Kernel 3: fused scores (WMMA) + streaming top-10 + softmax + gather + MLP.
// Block = 16 query rows, 8 waves. Each 16x16 score tile covers one stock's
// 16 lags (diag mask = one uniform compare). Per-lane register top-10
// (2 lanes per row per wave), merged per row by 16 scalar threads.
// ---------------------------------------------------------------------------
__global__ __launch_bounds__(256) void DeltaLag_score_topk(
    const float* __restrict__ queries,  // (N,H)
    const float* __restrict__ keys,     // (N*L,H)
    const float* __restrict__ x,        // (N,T,F)
    const float* __restrict__ W1, const float* __restrict__ b1,
    const float* __restrict__ W2, const float* __restrict__ b2,
    float* __restrict__ out)            // (N,)
{
  __shared__ float stile[8][16 * 16];
  __shared__ float cval[8][32][TOPK_K];
  __shared__ int   cidx[8][32][TOPK_K];

  const int tid  = threadIdx.x;
  const int wave = tid >> 5;
  const int lane = tid & 31;
  const int m = lane & 15, khi = lane >> 4;
  const int n0 = blockIdx.x * 16;

  // A fragments (queries tile) are loop-invariant -> registers
  v2f qf[16];
#pragma unroll
  for (int s = 0; s < 16; ++s) {
    int kk = 4 * s + 2 * khi;
    qf[s].x = queries[(size_t)(n0 + m) * H_DIM + kk];
    qf[s].y = queries[(size_t)(n0 + m) * H_DIM + kk + 1];
  }

  float tv[TOPK_K]; int ti[TOPK_K];
#pragma unroll
  for (int k = 0; k < TOPK_K; ++k) { tv[k] = -3.0e38f; ti[k] = 0; }
  float curMin = -3.0e38f;

  const int srow  = lane & 15;        // scan row
  const int cbase = (lane >> 4) * 8;  // scan col half

#pragma unroll 1
  for (int ct = wave; ct < N_STOCKS; ct += 8) {
    v8f acc = {};
    const float* kbase = keys + (size_t)ct * (16 * H_DIM);
#pragma unroll
    for (int s = 0; s < 16; ++s) {
      int kk = 4 * s + 2 * khi;
      v2f b; b.x = kbase[m * H_DIM + kk]; b.y = kbase[m * H_DIM + kk + 1];
      acc = wmma4(qf[s], b, acc);
    }
#pragma unroll
    for (int r = 0; r < 8; ++r)
      stile[wave][(r + 8 * khi) * 16 + m] = acc[r];
    asm volatile("s_wait_dscnt 0" ::: "memory");  // intra-wave LDS RAW
#pragma unroll
    for (int cc = 0; cc < 8; ++cc) {
      int col = cbase + cc;
      float v = stile[wave][srow * 16 + col];
      if (ct == n0 + srow) v = -1.0e9f;           // diagonal mask
      if (v > curMin) {
        int slot = 0; float mn = tv[0];
#pragma unroll
        for (int k = 1; k < TOPK_K; ++k) if (tv[k] < mn) { mn = tv[k]; slot = k; }
        tv[slot] = v; ti[slot] = ct * 16 + col;
        mn = tv[0];
#pragma unroll
        for (int k = 1; k < TOPK_K; ++k) mn = fminf(mn, tv[k]);
        curMin = mn;
      }
    }
    asm volatile("" ::: "memory");
  }

#pragma unroll
  for (int k = 0; k < TOPK_K; ++k) {
    cval[wave][lane][k] = tv[k];
    cidx[wave][lane][k] = ti[k];
  }
  __syncthreads();

  if (tid < 16) {
    const int row = tid;
    float bv[TOPK_K]; int bi[TOPK_K];
#pragma unroll
    for (int k = 0; k < TOPK_K; ++k) { bv[k] = -3.0e38f; bi[k] = 0; }
    float mn2 = -3.0e38f;
    for (int w = 0; w < 8; ++w) {
      for (int half = 0; half < 2; ++half) {
        int ln = row + 16 * half;
        for (int k = 0; k < TOPK_K; ++k) {
          float v = cval[w][ln][k];
          if (v > mn2) {
            int slot = 0; float mn = bv[0];
            for (int q = 1; q < TOPK_K; ++q) if (bv[q] < mn) { mn = bv[q]; slot = q; }
            bv[slot] = v; bi[slot] = cidx[w][ln][k];
            mn = bv[0];
            for (int q = 1; q < TOPK_K; ++q) mn = fminf(mn, bv[q]);
            mn2 = mn;
          }
        }
      }
    }
    // softmax over top-10 (permutation invariant)
    float mx = bv[0];
    for (int k = 1; k < TOPK_K; ++k) mx = fmaxf(mx, bv[k]);
    float ew[TOPK_K], wsum = 0.0f;
    for (int k = 0; k < TOPK_K; ++k) { ew[k] = __expf(bv[k] - mx); wsum += ew[k]; }
    float inv = 1.0f / wsum;
    // gather leader features, weighted sum
    float feat[F_DIM];
    for (int f = 0; f < F_DIM; ++f) feat[f] = 0.0f;
    for (int k = 0; k < TOPK_K; ++k) {
      int gi = bi[k];
      int leader = gi >> 4;
      int lag = gi & 15;
      int tix = T_STEPS - (L_LAGS - lag);
      const float* xp = x + ((size_t)leader * T_STEPS + tix) * F_DIM;
      float wk = ew[k] * inv;
      for (int f = 0; f < F_DIM; ++f) feat[f] += wk * xp[f];
    }
    // MLP: leaky_relu(feat @ W1^T + b1) @ W2^T + b2
    float o = b2[0];
    for (int j = 0; j < F_DIM; ++j) {
      float hj = b1[j];
      for (int f = 0; f < F_DIM; ++f) hj += W1[j * F_DIM + f] * feat[f];
      hj = (hj > 0.0f) ? hj : 0.01f * hj;
      o += W2[j] * hj;
    }
    out[n0 + row] = o;
  }
}

// ---------------------------------------------------------------------------
extern "C" void kernel_launch(void* const* d_in, const int* in_sizes, int n_in,
                              void* d_out, int out_size, void* d_ws, size_t ws_size,
                              hipStream_t stream) {
  (void)in_sizes; (void)n_in; (void)out_size; (void)ws_size;
  const float* x    = (const float*)d_in[0];
  const float* W_ih = (const float*)d_in[1];
  const float* W_hh = (const float*)d_in[2];
  const float* b_ih = (const float*)d_in[3];
  const float* b_hh = (const float*)d_in[4];
  const float* Wq   = (const float*)d_in[5];
  const float* Wk   = (const float*)d_in[6];
  const float* W1   = (const float*)d_in[7];
  const float* b1   = (const float*)d_in[8];
  const float* W2   = (const float*)d_in[9];
  const float* b2   = (const float*)d_in[10];
  float* out = (float*)d_out;

  // workspace layout (floats): hkeys (N*L*H) | keys (N*L*H) | queries (N*H)
  float* hkeys   = (float*)d_ws;
  float* keysbuf = hkeys + (size_t)N_STOCKS * L_LAGS * H_DIM;
  float* queries = keysbuf + (size_t)N_STOCKS * L_LAGS * H_DIM;

  DeltaLag_gru<<<N_STOCKS / 16, 128, 0, stream>>>(x, W_ih, W_hh, b_ih, b_hh, hkeys);
  // keys = hkeys @ Wk^T  (32768 rows)
  DeltaLag_proj<<<(N_STOCKS * L_LAGS) / 16, 32, 0, stream>>>(hkeys, H_DIM, Wk, keysbuf);
  // queries = h[:,T-1] @ Wq^T  (stride L*H picks lag 15 of each sample)
  DeltaLag_proj<<<N_STOCKS / 16, 32, 0, stream>>>(hkeys + (L_LAGS - 1) * H_DIM,
                                                  L_LAGS * H_DIM, Wq, queries);
  DeltaLag_score_topk<<<N_STOCKS / 16, 256, 0, stream>>>(queries, keysbuf, x,
                                                         W1, b1, W2, b2, out);
}